// FanoCoupling_43224550867196
// MI455X (gfx1250) — compile-verified
//
#include <hip/hip_runtime.h>

typedef float v2f __attribute__((ext_vector_type(2)));
typedef float v4f __attribute__((ext_vector_type(4)));
typedef float v8f __attribute__((ext_vector_type(8)));
typedef unsigned int uint32x4 __attribute__((ext_vector_type(4)));
typedef int int32x4 __attribute__((ext_vector_type(4)));
typedef int int32x8 __attribute__((ext_vector_type(8)));

#define ZDIM 64
#define NCOL 7
#define ROWSTRIDE (NCOL * ZDIM)   /* 448 floats per batch row */
#define TILE 16                   /* batch rows per block     */

/* TDM pads 4 DWORDs after every 64 DWORDs -> element e lands at e + 4*(e/64).
   Row stride = 448 + 28 = 476 dwords, colony stride = 64 + 4 = 68 dwords. */
#define STROW 476
#define STCOL 68
#define STSIZE (TILE * STROW)     /* 7616 floats */

#define LDSROW 65                 /* padded delta row stride (floats) */
#define LDSLINE (TILE * LDSROW)

// Fano tables packed as nibbles (index l -> (packed >> 4l) & 0xF)
#define PACK_I 0x4341000u  /* I = {0,0,0,1,4,3,4} */
#define PACK_J 0x2213531u  /* J = {1,3,5,3,1,2,2} */

#define USE_TDM (__has_builtin(__builtin_amdgcn_tensor_load_to_lds) && \
                 __has_builtin(__builtin_amdgcn_s_wait_tensorcnt))

__global__ __launch_bounds__(256) void fano_wmma_kernel(
    const float* __restrict__ states, const float* __restrict__ W,
    const float* __restrict__ bias, const float* __restrict__ gates,
    float* __restrict__ out)
{
    __shared__ float stTile[STSIZE];          /* padded state tile [16][7][64] */
    __shared__ float ldsDelta[NCOL * LDSLINE];

    const int tid  = threadIdx.x;
    const int wave = tid >> 5;
    const int lane = tid & 31;
    const int half = lane >> 4;
    const int l16  = lane & 15;
    const long tileBase = (long)blockIdx.x * TILE;
    const float* tileSrc = states + tileBase * ROWSTRIDE;

#if USE_TDM
    // ---- wave 7: one TDM descriptor DMAs the whole 28KB tile into LDS ----
    if (wave == NCOL) {
        unsigned long long ga = (unsigned long long)(const void*)tileSrc;
        unsigned int ldsAddr  = (unsigned int)(unsigned long long)(void*)stTile;
        uint32x4 g0;
        g0.x = 1u;                                       // count=1 (valid, user mode)
        g0.y = ldsAddr;                                  // lds_addr [63:32]
        g0.z = (unsigned int)(ga & 0xFFFFFFFFu);         // global_addr lo
        g0.w = (unsigned int)((ga >> 32) & 0x01FFFFFFu)  // global_addr [56:32]
             | (2u << 30);                               // type=2 ("image")
        int32x8 g1;
        g1[0] = (2 << 16)      // data_size = 4 bytes
              | (1 << 20)      // pad_enable
              | (5 << 22)      // pad_interval: code 5 = every 64 DWORDs
              | (3 << 25);     // pad_amount:   code 3 = 4 DWORDs
        g1[1] = (int)(448u << 16);   // tensor_dim0[15:0] = 448  (bits 79:48)
        g1[2] = (int)(16u  << 16);   // tensor_dim1[15:0] = 16   (bits 111:80)
        g1[3] = (int)(448u << 16);   // tile_dim0 = 448          (bits 127:112)
        g1[4] = 16;                  // tile_dim1 = 16, tile_dim2 = 0
        g1[5] = 448;                 // tensor_dim0_stride = 448 (contiguous rows)
        g1[6] = 0;
        g1[7] = 0;
        int32x4 z4 = {0, 0, 0, 0};
#if __clang_major__ >= 23
        int32x8 z8 = {0, 0, 0, 0, 0, 0, 0, 0};
        __builtin_amdgcn_tensor_load_to_lds(g0, g1, z4, z4, z8, 0);
#else
        __builtin_amdgcn_tensor_load_to_lds(g0, g1, z4, z4, 0);
#endif
    }
#else
    // ---- fallback: all 256 threads stage the tile coalesced (same layout) ----
#pragma unroll
    for (int k = 0; k < (TILE * ROWSTRIDE) / (4 * 256); ++k) {   // 7 iters
        const int f4 = tid + k * 256;
        const int e  = f4 * 4;
        v4f v = *(const v4f*)(tileSrc + e);
        *(v4f*)(stTile + e + 4 * (e >> 6)) = v;
    }
#endif

    // ---- softmax over the 7 gates (tiny; redundant per thread) ----
    float g[NCOL];
    float mx = gates[0];
#pragma unroll
    for (int i = 1; i < NCOL; ++i) mx = fmaxf(mx, gates[i]);
    float ssum = 0.f;
#pragma unroll
    for (int i = 0; i < NCOL; ++i) { g[i] = __expf(gates[i] - mx); ssum += g[i]; }
    const float inv = 1.0f / ssum;
#pragma unroll
    for (int i = 0; i < NCOL; ++i) g[i] *= inv;

#if USE_TDM
    if (wave == NCOL) __builtin_amdgcn_s_wait_tensorcnt(0);
#endif
    __syncthreads();   // state tile visible to all waves

    // ---- GEMM stage: waves 0..6 each compute one Fano line's delta[16,64] ----
    if (wave < NCOL) {
        const int line = wave;
        const int ci = (PACK_I >> (4 * line)) & 0xF;
        const int cj = (PACK_J >> (4 * line)) & 0xF;

        // A (16x4 f32): lane M = lane&15; VGPR0/1 hold K = 2*half, 2*half+1.
        // Read from the TDM-padded LDS tile: bank-conflict-free by construction.
        const float* aI = stTile + l16 * STROW + ci * STCOL + 2 * half;
        const float* aJ = stTile + l16 * STROW + cj * STCOL + 2 * half;
        // B (4x16 f32): lane N = lane&15; B[k][n] = W[n][k], W row-major [64x128]
        const float* w0 = W + (long)( 0 + l16) * 128 + 2 * half;
        const float* w1 = W + (long)(16 + l16) * 128 + 2 * half;
        const float* w2 = W + (long)(32 + l16) * 128 + 2 * half;
        const float* w3 = W + (long)(48 + l16) * 128 + 2 * half;

        v8f acc0 = {}, acc1 = {}, acc2 = {}, acc3 = {};
#pragma unroll
        for (int ks = 0; ks < 32; ++ks) {          // K = 128, 4 per WMMA
            const float* ap = (ks < 16) ? (aI + 4 * ks) : (aJ + 4 * (ks - 16));
            v2f a  = *(const v2f*)ap;
            v2f b0 = *(const v2f*)(w0 + 4 * ks);
            v2f b1 = *(const v2f*)(w1 + 4 * ks);
            v2f b2 = *(const v2f*)(w2 + 4 * ks);
            v2f b3 = *(const v2f*)(w3 + 4 * ks);
            acc0 = __builtin_amdgcn_wmma_f32_16x16x4_f32(false, a, false, b0, (short)0, acc0, false, false);
            acc1 = __builtin_amdgcn_wmma_f32_16x16x4_f32(false, a, false, b1, (short)0, acc1, false, false);
            acc2 = __builtin_amdgcn_wmma_f32_16x16x4_f32(false, a, false, b2, (short)0, acc2, false, false);
            acc3 = __builtin_amdgcn_wmma_f32_16x16x4_f32(false, a, false, b3, (short)0, acc3, false, false);
        }

        float gl = 0.f;
#pragma unroll
        for (int i = 0; i < NCOL; ++i) if (line == i) gl = g[i];

        // D layout: VGPR v -> M = v + 8*half, N = nt*16 + (lane&15)
        float* dl = ldsDelta + line * LDSLINE;
#pragma unroll
        for (int v = 0; v < 8; ++v) {
            const int M = v + 8 * half;
            dl[M * LDSROW +  0 + l16] = gl * acc0[v];
            dl[M * LDSROW + 16 + l16] = gl * acc1[v];
            dl[M * LDSROW + 32 + l16] = gl * acc2[v];
            dl[M * LDSROW + 48 + l16] = gl * acc3[v];
        }
    }
    __syncthreads();

    // ---- combine: out[b,c,z] = S[b,c,z] + coef[c]*bias[z] + sum(line deltas) ----
    constexpr int cK[NCOL] = {2, 4, 6, 5, 6, 6, 5};   // scatter targets (duplicates!)
    float* dstBase = out + tileBase * ROWSTRIDE;
#pragma unroll
    for (int it = 0; it < (TILE * ROWSTRIDE) / 256; ++it) {   // 28 iterations
        const int flat = it * 256 + tid;
        const int b   = flat / ROWSTRIDE;
        const int rem = flat - b * ROWSTRIDE;
        const int c   = rem >> 6;
        const int z   = rem & 63;
        float val = stTile[b * STROW + c * STCOL + z];   // state from LDS tile
        float coef = 0.f, dsum = 0.f;
#pragma unroll
        for (int l = 0; l < NCOL; ++l) {
            if (cK[l] == c) {
                coef += g[l];
                dsum += ldsDelta[l * LDSLINE + b * LDSROW + z];
            }
        }
        dstBase[flat] = val + coef * bias[z] + dsum;
    }
}

extern "C" void kernel_launch(void* const* d_in, const int* in_sizes, int n_in,
                              void* d_out, int out_size, void* d_ws, size_t ws_size,
                              hipStream_t stream) {
    (void)n_in; (void)out_size; (void)d_ws; (void)ws_size;
    const float* states = (const float*)d_in[0];
    const float* W      = (const float*)d_in[1];
    const float* bias   = (const float*)d_in[2];
    const float* gates  = (const float*)d_in[3];
    float* out          = (float*)d_out;

    const int B = in_sizes[0] / ROWSTRIDE;   // 131072
    const int blocks = B / TILE;             // 8192
    fano_wmma_kernel<<<blocks, 256, 0, stream>>>(states, W, bias, gates, out);
}